// CostVolumeBidirectional_21114059227574
// MI455X (gfx1250) — compile-verified
//
#include <hip/hip_runtime.h>

// Cost volume via V_WMMA_F32_16X16X4_F32 (CDNA5 / gfx1250, wave32).
//
// Memory-bound: ~190 MB effective traffic vs 0.42 GFLOP useful work
// (AI ~ 2 FLOP/B; HBM roofline ~8.5 us @ 23.3 TB/s). Keep fp32 (matches
// reference numerics, same bandwidth) and do the 96-deep channel dot
// products as 24 chained K=4 fp32 WMMAs per 16x16 Gram tile.
//
// v2: branchless hot loop. Boundary columns are loaded from a clamped
// in-bounds address; the zeroing required for the reference's output
// padding is folded into the per-column store scale (valid ? 1/96 : 0),
// eliminating the per-iteration EXEC save/restore seen in round 1.

typedef __attribute__((ext_vector_type(2))) float v2f;
typedef __attribute__((ext_vector_type(8))) float v8f;

#define BS 8
#define FS 96
#define HH 96
#define WW 320
#define PLANES 9           // 1 + 2*4
#define LVLS 4
#define TILES_X (WW / 16)  // 20
#define NWAVES (BS * HH * TILES_X)  // 15360
#define WAVES_PER_BLOCK 8

__global__ __launch_bounds__(256) void cost_volume_wmma_f32(
    const float* __restrict__ f1, const float* __restrict__ f2,
    float* __restrict__ out) {
  const int lane = threadIdx.x & 31;
  const int wid = blockIdx.x * WAVES_PER_BLOCK + (threadIdx.x >> 5);

  const int t = wid % TILES_X;        // 16-column row-tile of f1
  const int bh = wid / TILES_X;
  const int h = bh % HH;
  const int b = bh / HH;

  const int X0 = t * 16;
  const int half = lane >> 4;         // 0: K={0,1}, 1: K={2,3}
  const int m = lane & 15;            // M for A, N for B

  const int HW = HH * WW;
  const float* __restrict__ f1row = f1 + ((long)b * FS * HH + h) * WW;
  const float* __restrict__ f2row = f2 + ((long)b * FS * HH + h) * WW;

  // Two B column groups covering band columns [X0-4, X0+20).
  // col0 in [-8, 311]: only the lower bound can be violated.
  // col1 in [  8, 327]: only the upper bound can be violated.
  const int col0 = X0 - 8 + m;
  const int col1 = X0 + 8 + m;
  const int col0c = col0 < 0 ? 0 : col0;            // clamped, in-bounds
  const int col1c = col1 >= WW ? (WW - 1) : col1;   // clamped, in-bounds
  const float inv = 1.0f / (float)FS;
  const float scale0 = (col0 >= 0) ? inv : 0.0f;    // folds /96 + pad-zeroing
  const float scale1 = (col1 < WW) ? inv : 0.0f;

  // Per-lane element indices (channel advances by 4 per k-step).
  const int aBase = 2 * half * HW + X0 + m;

  v8f acc0 = {};
  v8f acc1 = {};

#pragma unroll
  for (int k0 = 0; k0 < FS; k0 += 4) {
    const int c = k0 * HW;            // + (2*half*HW) already in bases

    // A fragment: A[M=m][K] = f1[k0 + 2*half + K][X0+m]
    v2f a;
    a.x = f1row[c + aBase];
    a.y = f1row[c + aBase + HW];

    // B fragments: B[K][N=m] = f2[k0 + 2*half + K][col]; unconditional,
    // clamped addresses (garbage values are zeroed at store time).
    v2f b0, b1;
    b0.x = f2row[c + 2 * half * HW + col0c];
    b0.y = f2row[c + 2 * half * HW + HW + col0c];
    b1.x = f2row[c + 2 * half * HW + col1c];
    b1.y = f2row[c + 2 * half * HW + HW + col1c];

    // 8-arg form: (neg_a, A, neg_b, B, c_mod, C, reuse_a, reuse_b)
    acc0 = __builtin_amdgcn_wmma_f32_16x16x4_f32(false, a, false, b0,
                                                 (short)0, acc0, false, false);
    acc1 = __builtin_amdgcn_wmma_f32_16x16x4_f32(false, a, false, b1,
                                                 (short)0, acc1, false, false);
  }

  // D layout: VGPR r holds (M = r + 8*half, N = m). Band/pad mapping:
  //   out[b][p = 4 - d][h][x],  d = x' - x = (col - X0) - M  (X0 cancels).
  // Each out element is written exactly once across all waves; invalid
  // x' columns store 0 via scale0/scale1, reproducing jnp.pad's zeros.
  const long outBase = ((long)b * PLANES * HH + h) * WW;
#pragma unroll
  for (int r = 0; r < 8; ++r) {
    const int M = r + 8 * half;
    const int x = X0 + M;

    const int d0 = m - 8 - M;         // col0 - x
    if (d0 >= -LVLS && d0 <= LVLS) {
      out[outBase + (long)(LVLS - d0) * HW + x] = acc0[r] * scale0;
    }
    const int d1 = m + 8 - M;         // col1 - x
    if (d1 >= -LVLS && d1 <= LVLS) {
      out[outBase + (long)(LVLS - d1) * HW + x] = acc1[r] * scale1;
    }
  }
}

extern "C" void kernel_launch(void* const* d_in, const int* in_sizes, int n_in,
                              void* d_out, int out_size, void* d_ws,
                              size_t ws_size, hipStream_t stream) {
  (void)in_sizes; (void)n_in; (void)d_ws; (void)ws_size; (void)out_size;
  const float* f1 = (const float*)d_in[0];
  const float* f2 = (const float*)d_in[1];
  float* out = (float*)d_out;

  const int blocks = NWAVES / WAVES_PER_BLOCK;  // 1920
  cost_volume_wmma_f32<<<dim3(blocks), dim3(256), 0, stream>>>(f1, f2, out);
}